// EdgeConv_hop_45174466019825
// MI455X (gfx1250) — compile-verified
//
#include <hip/hip_runtime.h>

typedef float v2f __attribute__((ext_vector_type(2)));
typedef float v8f __attribute__((ext_vector_type(8)));

#define C_DIM   128
#define NNODES  40000
#define NEDGES  640000
#define BN_EPS  1e-5f

#define K1_BLOCKS        160
#define K2_BLOCKS        512
#define WAVES_PER_BLOCK  8
#define NWAVES_K2        (K2_BLOCKS * WAVES_PER_BLOCK)

// Stage weight [K][N] row-major -> pair-interleaved LDS image:
//   wl[(k>>1)*256 + 2*n + (k&1)] = w[k*128 + n]
// so a lane's B fragment {w[k][n], w[k+1][n]} (k always even in the K-loop)
// is one aligned float2 -> single ds_load_b64, no repack moves.
__device__ __forceinline__ void stage_weight_paired(float* wl, const float* w, int tid) {
    for (int idx = tid; idx < C_DIM * C_DIM; idx += 256) {
        const int k = idx >> 7;
        const int n = idx & (C_DIM - 1);
        wl[(k >> 1) * 256 + 2 * n + (k & 1)] = w[idx];
    }
}

__device__ __forceinline__ v2f bfrag(const float* wl, int k, int n) {
    const float2 t = *(const float2*)&wl[(k >> 1) * 256 + 2 * n];
    v2f b; b.x = t.x; b.y = t.y; return b;
}

// ---------------------------------------------------------------------------
// K1: xh = x @ w_h ; xt = x @ w_t   (blockIdx.y selects which)
// ---------------------------------------------------------------------------
__global__ void __launch_bounds__(256) node_gemm(const float* __restrict__ x,
                                                 const float* __restrict__ w_h,
                                                 const float* __restrict__ w_t,
                                                 float* __restrict__ xh,
                                                 float* __restrict__ xt) {
    __shared__ float wl[C_DIM * C_DIM];
    const float* w   = (blockIdx.y == 0) ? w_h : w_t;
    float*       dst = (blockIdx.y == 0) ? xh : xt;
    stage_weight_paired(wl, w, threadIdx.x);
    __syncthreads();

    const int lane = threadIdx.x & 31;
    const int wave = threadIdx.x >> 5;
    const int ln   = lane & 15;          // row (A) / col (B,C,D) within tile
    const int half = lane >> 4;          // 0: K0..1/M0..7 ; 1: K2..3/M8..15
    const int kh   = half << 1;

    const int gw     = blockIdx.x * WAVES_PER_BLOCK + wave;
    const int nwaves = gridDim.x * WAVES_PER_BLOCK;
    const int ntiles = NNODES / 16;

    for (int t = gw; t < ntiles; t += nwaves) {
        const int r0 = t * 16;
        v8f acc[8];
#pragma unroll
        for (int nt = 0; nt < 8; ++nt) acc[nt] = (v8f){0.f,0.f,0.f,0.f,0.f,0.f,0.f,0.f};

#pragma unroll 4
        for (int k0 = 0; k0 < C_DIM; k0 += 4) {
            const float2 av = *(const float2*)(x + (size_t)(r0 + ln) * C_DIM + k0 + kh);
            v2f a; a.x = av.x; a.y = av.y;
#pragma unroll
            for (int nt = 0; nt < 8; ++nt) {
                acc[nt] = __builtin_amdgcn_wmma_f32_16x16x4_f32(
                    false, a, false, bfrag(wl, k0 + kh, nt * 16 + ln),
                    (short)0, acc[nt], false, false);
            }
        }
#pragma unroll
        for (int nt = 0; nt < 8; ++nt)
#pragma unroll
            for (int r = 0; r < 8; ++r)
                dst[(size_t)(r0 + r + 8 * half) * C_DIM + nt * 16 + ln] = acc[nt][r];
    }
}

// ---------------------------------------------------------------------------
// K2: out0 = edge_attr @ w_self ; fuse gathered xh/xt, residual;
//     write pre-BN output to d_out and per-wave BN partials to ws.
// ---------------------------------------------------------------------------
__global__ void __launch_bounds__(256) edge_gemm(const float* __restrict__ edge_attr,
                                                 const int*   __restrict__ eidx,
                                                 const float* __restrict__ w_self,
                                                 const float* __restrict__ xh,
                                                 const float* __restrict__ xt,
                                                 float* __restrict__ out,
                                                 float* __restrict__ psum,
                                                 float* __restrict__ psq) {
    __shared__ float wl[C_DIM * C_DIM];
    stage_weight_paired(wl, w_self, threadIdx.x);
    __syncthreads();

    const int lane = threadIdx.x & 31;
    const int wave = threadIdx.x >> 5;
    const int ln   = lane & 15;
    const int half = lane >> 4;
    const int kh   = half << 1;
    const int gw   = blockIdx.x * WAVES_PER_BLOCK + wave;

    float lsum[8], lsq[8];
#pragma unroll
    for (int nt = 0; nt < 8; ++nt) { lsum[nt] = 0.f; lsq[nt] = 0.f; }

    const int ntiles = NEDGES / 16;
    for (int t = gw; t < ntiles; t += NWAVES_K2) {
        const int e0 = t * 16;

        // prefetch next tile of the edge_attr HBM stream (global_prefetch_b8)
        const int tn = t + NWAVES_K2;
        if (tn < ntiles) {
            const float* pn = edge_attr + (size_t)(tn * 16 + ln) * C_DIM + half * 64;
            __builtin_prefetch(pn, 0, 1);
            __builtin_prefetch(pn + 32, 0, 1);
        }

        int ri[8], ci[8];
#pragma unroll
        for (int r = 0; r < 8; ++r) {
            const int m = e0 + r + 8 * half;
            ri[r] = eidx[m];
            ci[r] = eidx[NEDGES + m];
        }

        v8f acc[8];
#pragma unroll
        for (int nt = 0; nt < 8; ++nt) acc[nt] = (v8f){0.f,0.f,0.f,0.f,0.f,0.f,0.f,0.f};

#pragma unroll 4
        for (int k0 = 0; k0 < C_DIM; k0 += 4) {
            const float2 av = *(const float2*)(edge_attr + (size_t)(e0 + ln) * C_DIM + k0 + kh);
            v2f a; a.x = av.x; a.y = av.y;
#pragma unroll
            for (int nt = 0; nt < 8; ++nt) {
                acc[nt] = __builtin_amdgcn_wmma_f32_16x16x4_f32(
                    false, a, false, bfrag(wl, k0 + kh, nt * 16 + ln),
                    (short)0, acc[nt], false, false);
            }
        }

        // Epilogue: out = out0*(1 + 0.5*(head+tail)) + edge_attr
#pragma unroll
        for (int nt = 0; nt < 8; ++nt) {
            const int c = nt * 16 + ln;
#pragma unroll
            for (int r = 0; r < 8; ++r) {
                const int m = e0 + r + 8 * half;
                float o  = acc[nt][r];
                const float hv = xh[(size_t)ri[r] * C_DIM + c];
                const float tv = xt[(size_t)ci[r] * C_DIM + c];
                const float ea = edge_attr[(size_t)m * C_DIM + c];
                o = o + 0.5f * (hv + tv) * o + ea;
                out[(size_t)m * C_DIM + c] = o;
                lsum[nt] += o;
                lsq[nt]  += o * o;
            }
        }
    }

    // reduce half-lanes (wave32): lane L and L+16 hold the same channel set
#pragma unroll
    for (int nt = 0; nt < 8; ++nt) {
        lsum[nt] += __shfl_xor(lsum[nt], 16, 32);
        lsq[nt]  += __shfl_xor(lsq[nt], 16, 32);
    }
    if (half == 0) {
#pragma unroll
        for (int nt = 0; nt < 8; ++nt) {
            psum[(size_t)gw * C_DIM + nt * 16 + ln] = lsum[nt];
            psq [(size_t)gw * C_DIM + nt * 16 + ln] = lsq[nt];
        }
    }
}

// ---------------------------------------------------------------------------
// K3: reduce per-wave partials -> fused BN scale/shift (f64 accumulation)
// ---------------------------------------------------------------------------
__global__ void __launch_bounds__(128) bn_stats(const float* __restrict__ psum,
                                                const float* __restrict__ psq,
                                                const float* __restrict__ gamma,
                                                const float* __restrict__ beta,
                                                float* __restrict__ scale,
                                                float* __restrict__ shift) {
    const int c = threadIdx.x;
    double s = 0.0, q = 0.0;
    for (int w = 0; w < NWAVES_K2; ++w) {
        s += (double)psum[(size_t)w * C_DIM + c];
        q += (double)psq [(size_t)w * C_DIM + c];
    }
    const float inv  = 1.0f / (float)NEDGES;
    const float mean = (float)s * inv;
    const float var  = (float)q * inv - mean * mean;
    const float is   = rsqrtf(var + BN_EPS);
    const float a    = gamma[c] * is;
    scale[c] = a;
    shift[c] = beta[c] - mean * a;
}

// ---------------------------------------------------------------------------
// K4: in-place  d_out = relu(d_out*scale[c] + shift[c]),  float4 vectorized
// ---------------------------------------------------------------------------
__global__ void __launch_bounds__(256) bn_apply(float* __restrict__ out,
                                                const float* __restrict__ scale,
                                                const float* __restrict__ shift) {
    __shared__ float sc[C_DIM], sh[C_DIM];
    if (threadIdx.x < C_DIM) {
        sc[threadIdx.x] = scale[threadIdx.x];
        sh[threadIdx.x] = shift[threadIdx.x];
    }
    __syncthreads();
    const size_t total = (size_t)NEDGES * C_DIM / 4;
    const size_t step  = (size_t)gridDim.x * blockDim.x;
    for (size_t i = (size_t)blockIdx.x * blockDim.x + threadIdx.x; i < total; i += step) {
        float4 v = ((float4*)out)[i];
        const int c = (int)((i * 4) & (C_DIM - 1));
        v.x = fmaxf(v.x * sc[c + 0] + sh[c + 0], 0.f);
        v.y = fmaxf(v.y * sc[c + 1] + sh[c + 1], 0.f);
        v.z = fmaxf(v.z * sc[c + 2] + sh[c + 2], 0.f);
        v.w = fmaxf(v.w * sc[c + 3] + sh[c + 3], 0.f);
        ((float4*)out)[i] = v;
    }
}

// ---------------------------------------------------------------------------
extern "C" void kernel_launch(void* const* d_in, const int* in_sizes, int n_in,
                              void* d_out, int out_size, void* d_ws, size_t ws_size,
                              hipStream_t stream) {
    (void)in_sizes; (void)n_in; (void)out_size; (void)ws_size;

    const float* x         = (const float*)d_in[0];
    const int*   edge_idx  = (const int*)  d_in[1];  // [2, E]
    const float* edge_attr = (const float*)d_in[2];
    /* d_in[3] = edge_type (unused) */
    const float* w_self    = (const float*)d_in[4];
    const float* w_h       = (const float*)d_in[5];
    const float* w_t       = (const float*)d_in[6];
    const float* gamma     = (const float*)d_in[7];
    const float* beta_bn   = (const float*)d_in[8];
    float*       out       = (float*)d_out;

    // workspace layout (floats)
    float* ws = (float*)d_ws;
    const size_t node_elems = (size_t)NNODES * C_DIM;       // 5,120,000
    float* xh    = ws;
    float* xt    = ws + node_elems;
    float* psum  = ws + 2 * node_elems;
    float* psq   = psum + (size_t)NWAVES_K2 * C_DIM;
    float* scale = psq  + (size_t)NWAVES_K2 * C_DIM;
    float* shift = scale + C_DIM;

    node_gemm<<<dim3(K1_BLOCKS, 2), 256, 0, stream>>>(x, w_h, w_t, xh, xt);
    edge_gemm<<<K2_BLOCKS, 256, 0, stream>>>(edge_attr, edge_idx, w_self,
                                             xh, xt, out, psum, psq);
    bn_stats<<<1, 128, 0, stream>>>(psum, psq, gamma, beta_bn, scale, shift);
    bn_apply<<<4096, 256, 0, stream>>>(out, scale, shift);
}